// GraphTransformer_43714177138801
// MI455X (gfx1250) — compile-verified
//
#include <hip/hip_runtime.h>
#include <hip/hip_bf16.h>

typedef __attribute__((ext_vector_type(2))) float v2f;
typedef __attribute__((ext_vector_type(8))) float v8f;

#define NN 100000
#define EE 1600000
#define CIN 128
#define COUT 64
#define PAD 4   // LDS row padding (floats): keeps 16B alignment, spreads banks

// ---------- float atomic-max via order-preserving uint encoding ----------
__device__ __forceinline__ unsigned enc_f32(float x) {
    unsigned u = __float_as_uint(x);
    return (u & 0x80000000u) ? ~u : (u | 0x80000000u);
}
__device__ __forceinline__ float dec_f32(unsigned u) {
    unsigned v = (u & 0x80000000u) ? (u & 0x7FFFFFFFu) : ~u;
    return __uint_as_float(v);
}

// ---------- fused QKVS GEMM: o{0..3}[N,64] = A[N,K] @ W{0..3}[K,64] + b{0..3} ----------
// 512 threads = 16 waves. A-tile (16 x K) staged once into LDS with coalesced
// float4 loads; each wave computes one 16x16 WMMA tile: wave>>2 selects which
// of the 4 outputs, wave&3 selects the 16-column group. fp32 WMMA (16x16x4).
// Requires N % 16 == 0 (100000 = 6250*16): no divergence at WMMA (EXEC all-1s).
__global__ void __launch_bounds__(512)
gemm_qkvs_kernel(const float* __restrict__ A,
                 const float* __restrict__ W0, const float* __restrict__ W1,
                 const float* __restrict__ W2, const float* __restrict__ W3,
                 const float* __restrict__ b0, const float* __restrict__ b1,
                 const float* __restrict__ b2, const float* __restrict__ b3,
                 float* __restrict__ o0, float* __restrict__ o1,
                 float* __restrict__ o2, float* __restrict__ o3,
                 int K)
{
    __shared__ float As[16 * (CIN + PAD)];
    const int t    = threadIdx.x;
    const int tile = blockIdx.x * 16;
    const int ldk  = K + PAD;

    // ---- stage A tile: 16*K floats, coalesced float4 ----
    const int nf4 = 16 * K / 4;                 // 512 (K=128) or 256 (K=64)
    const int kf4 = K / 4;
    for (int i = t; i < nf4; i += 512) {
        float4 val = ((const float4*)(A + (size_t)tile * K))[i];
        int r = i / kf4;
        int c = (i - r * kf4) * 4;
        *(float4*)(As + r * ldk + c) = val;
    }
    __syncthreads();

    const int wave = t >> 5;
    const int lane = t & 31;
    const int half = lane >> 4;                 // 0: lanes 0-15, 1: lanes 16-31
    const int l16  = lane & 15;
    const int oi   = wave >> 2;                 // which output (q,k,v,s)
    const int col0 = (wave & 3) * 16;           // 16-column group
    const int koff = half * 2;                  // lanes 16-31 hold K+2,K+3

    const float* W = (oi == 0) ? W0 : (oi == 1) ? W1 : (oi == 2) ? W2 : W3;
    const float* b = (oi == 0) ? b0 : (oi == 1) ? b1 : (oi == 2) ? b2 : b3;
    float*       o = (oi == 0) ? o0 : (oi == 1) ? o1 : (oi == 2) ? o2 : o3;

    const float* arow = As + l16 * ldk + koff;  // A fragment base (LDS, b64 reads)
    v8f acc = {};
    for (int kk = 0; kk < K; kk += 4) {
        v2f a, bm;
        a.x  = arow[kk];
        a.y  = arow[kk + 1];
        bm.x = W[(kk + koff) * COUT + col0 + l16];
        bm.y = W[(kk + koff + 1) * COUT + col0 + l16];
        acc = __builtin_amdgcn_wmma_f32_16x16x4_f32(false, a, false, bm,
                                                    (short)0, acc, false, false);
    }
    // C/D layout: VGPR i -> M = i (lanes 0-15) or M = 8+i (lanes 16-31)
    const int orow = tile + half * 8;
    const int col  = col0 + l16;
    const float bias = b[col];
#pragma unroll
    for (int i = 0; i < 8; ++i)
        o[(orow + i) * COUT + col] = acc[i] + bias;
}

// ---------- per-edge attention logit + segment max ----------
__global__ void __launch_bounds__(256)
edge_scores_kernel(const int* __restrict__ ei, const float* __restrict__ q,
                   const float* __restrict__ k, float* __restrict__ scores,
                   unsigned* __restrict__ segmax)
{
    int e = blockIdx.x * blockDim.x + threadIdx.x;
    if (e >= EE) return;
    int s = ei[e];
    int d = ei[EE + e];
    const float4* qd = (const float4*)(q + (size_t)d * COUT);
    const float4* ks = (const float4*)(k + (size_t)s * COUT);
    float acc = 0.f;
#pragma unroll
    for (int i = 0; i < COUT / 4; ++i) {
        float4 a = qd[i], bb = ks[i];
        acc += a.x * bb.x + a.y * bb.y + a.z * bb.z + a.w * bb.w;
    }
    acc *= 0.125f;  // 1/sqrt(64)
    scores[e] = acc;
    atomicMax(segmax + d, enc_f32(acc));
}

// ---------- exp(score - segmax[dst]) + segment sum ----------
__global__ void __launch_bounds__(256)
edge_exp_kernel(const int* __restrict__ ei, float* __restrict__ scores,
                const unsigned* __restrict__ segmax, float* __restrict__ segsum)
{
    int e = blockIdx.x * blockDim.x + threadIdx.x;
    if (e >= EE) return;
    int d = ei[EE + e];
    float m = dec_f32(segmax[d]);
    float ex = __expf(scores[e] - m);
    scores[e] = ex;
    atomicAdd(segsum + d, ex);
}

// ---------- scatter alpha * v[src] into agg[dst]  (4 threads per edge) ----------
__global__ void __launch_bounds__(256)
edge_scatter_kernel(const int* __restrict__ ei, const float* __restrict__ ex,
                    const float* __restrict__ segsum, const float* __restrict__ v,
                    float* __restrict__ agg)
{
    long long t = (long long)blockIdx.x * blockDim.x + threadIdx.x;
    int e = (int)(t >> 2);
    int sub = (int)(t & 3);
    if (e >= EE) return;
    int s = ei[e];
    int d = ei[EE + e];
    float alpha = ex[e] / fmaxf(segsum[d], 1e-16f);
    const float* vs = v + (size_t)s * COUT + sub * 16;
    float* ad = agg + (size_t)d * COUT + sub * 16;
#pragma unroll
    for (int c = 0; c < 16; ++c)
        atomicAdd(ad + c, alpha * vs[c]);
}

// ---------- elementwise ReLU ----------
__global__ void __launch_bounds__(256)
relu_kernel(float* __restrict__ x, int n)
{
    int i = blockIdx.x * blockDim.x + threadIdx.x;
    if (i < n) x[i] = fmaxf(x[i], 0.f);
}

extern "C" void kernel_launch(void* const* d_in, const int* in_sizes, int n_in,
                              void* d_out, int out_size, void* d_ws, size_t ws_size,
                              hipStream_t stream)
{
    const float* x   = (const float*)d_in[0];
    const int*   ei  = (const int*)d_in[1];
    // d_in[2] = edge_attr (ignored: edge_dim=None)
    const float* Wq1 = (const float*)d_in[3];  const float* bq1 = (const float*)d_in[4];
    const float* Wk1 = (const float*)d_in[5];  const float* bk1 = (const float*)d_in[6];
    const float* Wv1 = (const float*)d_in[7];  const float* bv1 = (const float*)d_in[8];
    const float* Ws1 = (const float*)d_in[9];  const float* bs1 = (const float*)d_in[10];
    const float* Wq2 = (const float*)d_in[11]; const float* bq2 = (const float*)d_in[12];
    const float* Wk2 = (const float*)d_in[13]; const float* bk2 = (const float*)d_in[14];
    const float* Wv2 = (const float*)d_in[15]; const float* bv2 = (const float*)d_in[16];
    const float* Ws2 = (const float*)d_in[17]; const float* bs2 = (const float*)d_in[18];
    float* out = (float*)d_out;

    // workspace carve-up (floats)
    float* q      = (float*)d_ws;
    float* k      = q + (size_t)NN * COUT;
    float* v      = k + (size_t)NN * COUT;
    float* h      = v + (size_t)NN * COUT;   // layer-1 agg (skip pre-loaded), then input to layer 2
    float* scores = h + (size_t)NN * COUT;   // E floats, reused as exp values
    float* segsum = scores + EE;             // N floats
    unsigned* segmax = (unsigned*)(segsum + NN);  // N uints

    const dim3 gB(512);
    const dim3 gGemm(NN / 16);               // 6250, exact
    const dim3 eB(256);
    const dim3 gE((EE + 255) / 256);         // 6250, exact
    const dim3 gE4((4LL * EE + 255) / 256);  // 25000, exact
    const dim3 gR((NN * COUT + 255) / 256);

    // ================= layer 1 =================
    gemm_qkvs_kernel<<<gGemm, gB, 0, stream>>>(x,
        Wq1, Wk1, Wv1, Ws1, bq1, bk1, bv1, bs1, q, k, v, h, CIN);  // s -> h (skip preload)
    hipMemsetAsync(segmax, 0, sizeof(unsigned) * NN, stream);
    hipMemsetAsync(segsum, 0, sizeof(float) * NN, stream);
    edge_scores_kernel<<<gE, eB, 0, stream>>>(ei, q, k, scores, segmax);
    edge_exp_kernel<<<gE, eB, 0, stream>>>(ei, scores, segmax, segsum);
    edge_scatter_kernel<<<gE4, eB, 0, stream>>>(ei, scores, segsum, v, h);
    relu_kernel<<<gR, eB, 0, stream>>>(h, NN * COUT);

    // ================= layer 2 =================
    gemm_qkvs_kernel<<<gGemm, gB, 0, stream>>>(h,
        Wq2, Wk2, Wv2, Ws2, bq2, bk2, bv2, bs2, q, k, v, out, COUT);  // s -> out (skip preload)
    hipMemsetAsync(segmax, 0, sizeof(unsigned) * NN, stream);
    hipMemsetAsync(segsum, 0, sizeof(float) * NN, stream);
    edge_scores_kernel<<<gE, eB, 0, stream>>>(ei, q, k, scores, segmax);
    edge_exp_kernel<<<gE, eB, 0, stream>>>(ei, scores, segmax, segsum);
    edge_scatter_kernel<<<gE4, eB, 0, stream>>>(ei, scores, segsum, v, out);
}